// VectorQuantizer_65017214927379
// MI455X (gfx1250) — compile-verified
//
#include <hip/hip_runtime.h>
#include <hip/hip_bf16.h>
#include <math.h>

// ---------------------------------------------------------------------------
// VQ forward for MI455X (gfx1250, wave32).
// S = sig + sig^T is symmetric -> cross = (xS)E^T = x (E S)^T.  x_quad and the
// per-row w scale never change the argmin, so the dominant work is ONE fused
// f32 GEMM (N x K x D = 34.4 GFLOP) + running argmin via V_WMMA_F32_16X16X4_F32.
// Memory floor ~192 MB @ 23.3 TB/s ~ 8 us; ES (1 MB) stays L2-resident.
// ---------------------------------------------------------------------------

typedef __attribute__((ext_vector_type(2))) float v2f;
typedef __attribute__((ext_vector_type(4))) float v4f;
typedef __attribute__((ext_vector_type(8))) float v8f;

#define K_EMB   1024
#define D_EMB   256
#define N_ROWS  65536
#define BT_STRIDE 260   // padded B-tile row: (4*n + k) mod 64 distinct -> conflict-free ds_load_b64

// ---------------- prep: S = sig + sig^T ----------------
__global__ __launch_bounds__(256) void k_build_S(const float* __restrict__ sig,
                                                 float* __restrict__ S) {
  int i = blockIdx.x, j = threadIdx.x;
  S[i * D_EMB + j] = sig[i * D_EMB + j] + sig[j * D_EMB + i];
}

// ---------------- prep: ES = E*S and e_quad[k] = <ES[k], E[k]> ----------------
__global__ __launch_bounds__(256) void k_es_equad(const float* __restrict__ E,
                                                  const float* __restrict__ S,
                                                  float* __restrict__ ES,
                                                  float* __restrict__ eq) {
  __shared__ float erow[D_EMB];
  __shared__ float red[256];
  int k = blockIdx.x, j = threadIdx.x;
  erow[j] = E[(size_t)k * D_EMB + j];
  __syncthreads();
  float acc = 0.0f;
  for (int i = 0; i < D_EMB; ++i) acc += erow[i] * S[i * D_EMB + j];
  ES[(size_t)k * D_EMB + j] = acc;
  red[j] = acc * erow[j];
  __syncthreads();
  for (int s = 128; s > 0; s >>= 1) { if (j < s) red[j] += red[j + s]; __syncthreads(); }
  if (j == 0) eq[k] = red[0];
}

// ---------------- zero histogram ----------------
__global__ void k_zero_counts(unsigned int* __restrict__ counts) {
  counts[blockIdx.x * 256 + threadIdx.x] = 0u;
}

// ---------------- main: fused cross-GEMM + argmin via WMMA f32 16x16x4 ----------------
// Block = 256 threads = 8 waves; each wave owns a 16-row A tile resident in VGPRs
// (fully unrolled accumulation -> constant indices -> no scratch).  B tiles are
// double-buffered in LDS; the NEXT tile's global loads are issued AFTER the
// barrier so they overlap the 64-WMMA chain (syncthreads implies wait-all, so
// issuing them before the barrier would expose the full load latency).
// B fragments are hand-pipelined two groups deep to keep LDS fetches ahead of
// the XDL pipe instead of a ds_load -> wait -> wmma lockstep.
__global__ __launch_bounds__(256, 2) void k_cross_argmin(
    const float* __restrict__ X, const float* __restrict__ ES,
    const float* __restrict__ eq, const float* __restrict__ W,
    int* __restrict__ out_idx, float* __restrict__ out_idx_f) {
  __shared__ float bt[2][16 * BT_STRIDE];

  const int tid  = threadIdx.x;
  const int lane = tid & 31;
  const int wave = tid >> 5;
  const int half = (lane < 16) ? 0 : 1;  // which K-pair this lane holds in A/B frags
  const int l16  = lane & 15;
  const int koff = half * 2;
  const int row_base = blockIdx.x * 128 + wave * 16;

  // A fragments: lane holds X[row_base + l16][4c + koff .. +1]  (ISA 16x4 f32 layout)
  v2f a[64];
  const float* xrow = X + (size_t)(row_base + l16) * D_EMB;
#pragma unroll
  for (int c = 0; c < 64; ++c) a[c] = *(const v2f*)(xrow + c * 4 + koff);

  float minv[8];
  int   mini[8];
#pragma unroll
  for (int r = 0; r < 8; ++r) { minv[r] = INFINITY; mini[r] = 0; }

  // ---- B-tile staging helpers (16x256 f32 = one 4096-float tile) ----
  v4f pf[4];
  auto tile_load = [&](int kt) {          // global -> regs, fully coalesced
    const float* src = ES + (size_t)kt * 16 * D_EMB;
#pragma unroll
    for (int i = 0; i < 4; ++i) pf[i] = *(const v4f*)(src + i * 1024 + tid * 4);
  };
  auto tile_store = [&](int buf) {        // regs -> padded LDS tile
#pragma unroll
    for (int i = 0; i < 4; ++i) {
      int e    = i * 1024 + tid * 4;
      int rrow = e >> 8;
      int col  = e & 255;
      *(v4f*)(&bt[buf][rrow * BT_STRIDE + col]) = pf[i];  // 16B-aligned
    }
  };

  tile_load(0);
  tile_store(0);

  for (int kt = 0; kt < 64; ++kt) {
    __syncthreads();                      // bt[kt&1] visible; prev reads of bt[(kt+1)&1] done
    if (kt + 1 < 64) tile_load(kt + 1);   // overlaps the WMMA chain below

    const float eqv  = eq[kt * 16 + l16];
    const int   colg = kt * 16 + l16;
    const float* btc = &bt[kt & 1][l16 * BT_STRIDE + koff];

    // two-deep pipelined B fragments: 4 frags (8 VGPRs) per group, 16 groups
    v2f bb[2][4];
#pragma unroll
    for (int j = 0; j < 4; ++j) bb[0][j] = *(const v2f*)(btc + j * 4);

    v8f c = {};
#pragma unroll                            // FULL unroll: constant indices -> regs
    for (int g = 0; g < 16; ++g) {
      if (g + 1 < 16) {
#pragma unroll
        for (int j = 0; j < 4; ++j)
          bb[(g + 1) & 1][j] = *(const v2f*)(btc + ((g + 1) * 4 + j) * 4);
      }
#pragma unroll
      for (int j = 0; j < 4; ++j)
        c = __builtin_amdgcn_wmma_f32_16x16x4_f32(false, a[g * 4 + j], false,
                                                  bb[g & 1][j], (short)0, c,
                                                  false, false);
    }

    // C layout: VGPR r -> row (r + 8*half), column l16 for this lane.
#pragma unroll
    for (int r = 0; r < 8; ++r) {
      float s = eqv - 2.0f * c[r];
      if (s < minv[r]) { minv[r] = s; mini[r] = colg; }  // strict < keeps lowest index
    }

    if (kt + 1 < 64) tile_store((kt + 1) & 1);  // consumes pf; ds stores before next barrier
  }

  // argmin merge across the 16 lanes of each half (columns n == lane (mod 16))
#pragma unroll
  for (int m = 1; m < 16; m <<= 1) {
#pragma unroll
    for (int r = 0; r < 8; ++r) {
      float ov = __shfl_xor(minv[r], m, 32);
      int   oi = __shfl_xor(mini[r], m, 32);
      if (ov < minv[r] || (ov == minv[r] && oi < mini[r])) { minv[r] = ov; mini[r] = oi; }
    }
  }

  if (l16 == 0) {  // lanes 0 and 16 write rows 0-7 / 8-15 of this wave's tile
#pragma unroll
    for (int r = 0; r < 8; ++r) {
      int row = row_base + half * 8 + r;
      float wv = W[row];
      int idx = (wv > 0.0f) ? mini[r] : 0;  // w==0 -> all-zero row -> jnp.argmin = 0
      out_idx[row]   = idx;
      out_idx_f[row] = (float)idx;
    }
  }
}

// ---------------- gather quantized, per-block MSE partials, histogram ----------------
__global__ __launch_bounds__(256) void k_gather_stats(
    const float* __restrict__ X, const float* __restrict__ E,
    const int* __restrict__ idx, float* __restrict__ qout,
    unsigned int* __restrict__ counts, float* __restrict__ partial) {
  __shared__ float red[256];
  int tid = threadIdx.x;
  int rl  = tid >> 5;   // row within block (8 rows/block)
  int ln  = tid & 31;   // 32 lanes per row
  int row = blockIdx.x * 8 + rl;
  int id  = idx[row];
  const float* e = E + (size_t)id  * D_EMB;
  const float* x = X + (size_t)row * D_EMB;
  float*       q = qout + (size_t)row * D_EMB;
  float acc = 0.0f;
#pragma unroll
  for (int j = 0; j < 2; ++j) {          // two coalesced 512B transactions per row
    int col = ln * 4 + j * 128;
    v4f ev = *(const v4f*)(e + col);
    v4f xv = *(const v4f*)(x + col);
    *(v4f*)(q + col) = ev;
    v4f d = ev - xv;
    acc += d.x * d.x + d.y * d.y + d.z * d.z + d.w * d.w;
  }
  if (ln == 0) atomicAdd(&counts[id], 1u);  // integer atomics: deterministic
  red[tid] = acc;
  __syncthreads();
  for (int s = 128; s > 0; s >>= 1) { if (tid < s) red[tid] += red[tid + s]; __syncthreads(); }
  if (tid == 0) partial[blockIdx.x] = red[0];
}

// ---------------- finalize: losses + perplexity (fixed-order reductions) ----------------
__global__ __launch_bounds__(256) void k_finalize(
    const float* __restrict__ W, const float* __restrict__ partial,
    const unsigned int* __restrict__ counts,
    float* __restrict__ out_scalars, float* __restrict__ out_perp) {
  __shared__ float red[256];
  int tid = threadIdx.x;

  float sw = 0.0f;
  for (int i = tid; i < N_ROWS; i += 256) sw += W[i];
  red[tid] = sw; __syncthreads();
  for (int s = 128; s > 0; s >>= 1) { if (tid < s) red[tid] += red[tid + s]; __syncthreads(); }
  float sumw = red[0]; __syncthreads();

  float ss = 0.0f;
  for (int i = tid; i < 8192; i += 256) ss += partial[i];
  red[tid] = ss; __syncthreads();
  for (int s = 128; s > 0; s >>= 1) { if (tid < s) red[tid] += red[tid + s]; __syncthreads(); }
  float sumsq = red[0]; __syncthreads();

  float ent = 0.0f;
  for (int i = tid; i < K_EMB; i += 256) {
    float p = (float)counts[i] / (float)N_ROWS;
    ent += p * logf(p + 1e-10f);
  }
  red[tid] = ent; __syncthreads();
  for (int s = 128; s > 0; s >>= 1) { if (tid < s) red[tid] += red[tid + s]; __syncthreads(); }

  if (tid == 0) {
    float mse = sumsq / (float)((size_t)N_ROWS * D_EMB);
    out_scalars[0] = mse * (sumw / (float)N_ROWS);  // codebook_loss
    out_scalars[1] = mse * 0.25f;                   // commitment_loss
    out_perp[0]    = expf(-red[0]);                 // perplexity
  }
}

// ---------------------------------------------------------------------------
extern "C" void kernel_launch(void* const* d_in, const int* in_sizes, int n_in,
                              void* d_out, int out_size, void* d_ws, size_t ws_size,
                              hipStream_t stream) {
  const float* X   = (const float*)d_in[0];  // [N, 256]
  const float* W   = (const float*)d_in[1];  // [N]
  const float* E   = (const float*)d_in[2];  // [1024, 256]
  const float* SIG = (const float*)d_in[3];  // [256, 256]

  float* out  = (float*)d_out;
  float* q    = out;                          // 16,777,216 quantized_st
  float* scal = out + 16777216;               // [codebook_loss, commitment_loss]
  float* idxf = out + 16777218;               // 65,536 indices (as float)
  float* perp = out + 16777218 + N_ROWS;      // perplexity

  float* ws = (float*)d_ws;                   // ~1.6 MB used
  float*        S       = ws;                               // 65536
  float*        ES      = ws + 65536;                       // 262144
  float*        eq      = ws + 65536 + 262144;              // 1024
  int*          idx     = (int*)(ws + 328704);              // 65536
  unsigned int* counts  = (unsigned int*)(ws + 394240);     // 1024
  float*        partial = ws + 395264;                      // 8192

  k_zero_counts<<<4, 256, 0, stream>>>(counts);
  k_build_S<<<256, 256, 0, stream>>>(SIG, S);
  k_es_equad<<<1024, 256, 0, stream>>>(E, S, ES, eq);
  k_cross_argmin<<<512, 256, 0, stream>>>(X, ES, eq, W, idx, idxf);
  k_gather_stats<<<8192, 256, 0, stream>>>(X, E, idx, q, counts, partial);
  k_finalize<<<1, 256, 0, stream>>>(W, partial, counts, scal, perp);
}